// TwoStreamSpatioTemporalGnn_47321949667506
// MI455X (gfx1250) — compile-verified
//
#include <hip/hip_runtime.h>

// ---------------------------------------------------------------------------
// Two-stream spatio-temporal GNN (DGCNN-style) forward for MI455X / gfx1250.
// f32 end-to-end; matrix ops use V_WMMA_F32_16X16X4_F32 (wave32 WMMA).
// ---------------------------------------------------------------------------

typedef __attribute__((ext_vector_type(2))) float v2f;
typedef __attribute__((ext_vector_type(8))) float v8f;

#define NNODES 8192
#define KNN    20
#define EPS_BN 1e-5f

// ---------------------------------------------------------------------------
// Input BatchNorm: biased stats over the 8192 rows for cols 0..2 (spatial)
// and col 3 (time). Produces per-column scale/shift in inss[0..3]/[4..7].
// ---------------------------------------------------------------------------
__global__ __launch_bounds__(256) void input_stats(
    const float* __restrict__ x,
    const float* __restrict__ sng, const float* __restrict__ snb,
    const float* __restrict__ tng, const float* __restrict__ tnb,
    float* __restrict__ inss)
{
    __shared__ float red[8][256];
    float s[4] = {0.f, 0.f, 0.f, 0.f}, s2[4] = {0.f, 0.f, 0.f, 0.f};
    for (int r = threadIdx.x; r < NNODES; r += 256) {
#pragma unroll
        for (int c = 0; c < 4; ++c) {
            float v = x[r * 4 + c];
            s[c] += v; s2[c] += v * v;
        }
    }
#pragma unroll
    for (int c = 0; c < 4; ++c) { red[c][threadIdx.x] = s[c]; red[4 + c][threadIdx.x] = s2[c]; }
    __syncthreads();
    if (threadIdx.x < 8) {
        float t = 0.f;
        for (int i = 0; i < 256; ++i) t += red[threadIdx.x][i];
        red[threadIdx.x][0] = t;
    }
    __syncthreads();
    if (threadIdx.x < 4) {
        const int c = threadIdx.x;
        const float m = red[c][0] * (1.0f / NNODES);
        const float v = red[4 + c][0] * (1.0f / NNODES) - m * m;
        const float g  = (c < 3) ? sng[c] : tng[0];
        const float be = (c < 3) ? snb[c] : tnb[0];
        const float a = g * rsqrtf(v + EPS_BN);
        inss[c] = a;
        inss[4 + c] = be - m * a;
    }
}

__global__ __launch_bounds__(256) void apply_norm(
    const float* __restrict__ x, const float* __restrict__ inss,
    float* __restrict__ XP)
{
    int t = blockIdx.x * 256 + threadIdx.x;       // 0 .. 8192*4
    int c = t & 3;
    XP[t] = x[t] * inss[c] + inss[4 + c];
}

// ---------------------------------------------------------------------------
// kNN (K=20) within each 2048-node batch segment. One 128-thread block per
// query node. dist = sq_i + sq_j - 2*dot (same formula as reference).
// Tie-break: lowest index (matches jax top_k stability). Includes self.
// ---------------------------------------------------------------------------
template <int DIM>
__global__ __launch_bounds__(128) void knn_kernel(
    const float* __restrict__ feat, int fstride, int* __restrict__ IDX)
{
    const int i   = blockIdx.x;
    const int tid = threadIdx.x;
    const int base = (i >> 11) << 11;             // batch segment start

    __shared__ float sfi[DIM];
    __shared__ float sdist[128];
    __shared__ int   sidx[128];

    for (int c = tid; c < DIM; c += 128) sfi[c] = feat[(size_t)i * fstride + c];
    __syncthreads();

    float fi[DIM];
#pragma unroll
    for (int c = 0; c < DIM; ++c) fi[c] = sfi[c];
    float sqi = 0.f;
#pragma unroll
    for (int c = 0; c < DIM; ++c) sqi += fi[c] * fi[c];

    float dl[16]; int jl[16];
#pragma unroll
    for (int s = 0; s < 16; ++s) {
        const int j = base + tid + s * 128;
        float dot = 0.f, sqj = 0.f;
#pragma unroll
        for (int c = 0; c < DIM; ++c) {
            float fj = feat[(size_t)j * fstride + c];
            dot += fi[c] * fj;
            sqj += fj * fj;
        }
        dl[s] = sqi + sqj - 2.0f * dot;
        jl[s] = j;
    }

    for (int r = 0; r < KNN; ++r) {
        float dmin = 1e30f; int jmin = 0x7fffffff;
#pragma unroll
        for (int s = 0; s < 16; ++s) {
            if (dl[s] < dmin || (dl[s] == dmin && jl[s] < jmin)) { dmin = dl[s]; jmin = jl[s]; }
        }
        sdist[tid] = dmin; sidx[tid] = jmin;
        __syncthreads();
        for (int off = 64; off > 0; off >>= 1) {
            if (tid < off) {
                float d2 = sdist[tid + off]; int j2 = sidx[tid + off];
                if (d2 < sdist[tid] || (d2 == sdist[tid] && j2 < sidx[tid])) {
                    sdist[tid] = d2; sidx[tid] = j2;
                }
            }
            __syncthreads();
        }
        const int wj = sidx[0];
        if (tid == 0) IDX[i * KNN + r] = wj;
        __syncthreads();
#pragma unroll
        for (int s = 0; s < 16; ++s) if (jl[s] == wj) dl[s] = 1e30f;
    }
}

// ---------------------------------------------------------------------------
// Weight pre-pad + K-pair interleave:
//   Wp[((k>>1)*dhid + col)*2 + (k&1)] = (k < din) ? W[k*dhid + col] : 0
// so a WMMA B-fragment (K = k0,k0+1 at fixed col, k0 even) is ONE b64 load.
// ---------------------------------------------------------------------------
__global__ __launch_bounds__(256) void pad_w(
    const float* __restrict__ W, int din, int dinpad, int dhid,
    float* __restrict__ Wp)
{
    const int t = blockIdx.x * 256 + threadIdx.x;
    if (t >= dinpad * dhid) return;
    const int k   = t / dhid;
    const int col = t - k * dhid;
    const float v = (k < din) ? W[(size_t)k * dhid + col] : 0.0f;
    Wp[((size_t)(k >> 1) * dhid + col) * 2 + (k & 1)] = v;
}

// ---------------------------------------------------------------------------
// Tiled f32 GEMM via V_WMMA_F32_16X16X4_F32.  Y[R x dhid] = A[R x din] @ W + b
// A is either gathered edge features [x_i, x_j - x_i] (idx != null) or a
// dense array (Adirect, row stride == DINPAD). One 16-row tile per wave,
// 4 waves/block; the A tile is staged zero-padded in LDS. ISA f32 layouts:
//   A 16x4 : lane L holds row M=L&15; vgpr{0,1} = K{0,1} (L<16) or K{2,3}.
//   B 4x16 : lane L holds col N=L&15; vgpr{0,1} = K{0,1} (L<16) or K{2,3}.
//   C 16x16: vgpr v -> row v + (L<16?0:8), col = L&15.
// ---------------------------------------------------------------------------
template <int DINPAD>
__global__ __launch_bounds__(128) void wmma_gemm(
    const float* __restrict__ feat, int fstride, int fdim,
    const int* __restrict__ idx,
    const float* __restrict__ Adirect,
    int din,
    const float* __restrict__ Wp, const float* __restrict__ bias,
    float* __restrict__ Y, int dhid)
{
    __shared__ float sh[4][16 * DINPAD];
    const int wave = threadIdx.x >> 5;
    const int lane = threadIdx.x & 31;
    const int tileBase = (blockIdx.x * 4 + wave) * 16;

    // Stage the 16 x DINPAD A-tile (zero-padded K) into LDS.
    for (int el = lane; el < 16 * DINPAD; el += 32) {
        const int row = el / DINPAD;
        const int c   = el - row * DINPAD;
        const int r   = tileBase + row;
        float v = 0.0f;
        if (idx != nullptr) {
            if (c < din) {
                const int n = r / KNN;
                const int k = r - n * KNN;
                const int j = idx[n * KNN + k];
                if (c < fdim) {
                    v = feat[(size_t)n * fstride + c];
                } else {
                    const int cc = c - fdim;
                    v = feat[(size_t)j * fstride + cc] - feat[(size_t)n * fstride + cc];
                }
            }
        } else {
            v = Adirect[(size_t)r * DINPAD + c];   // direct mode: din == DINPAD
        }
        sh[wave][el] = v;
    }
    __syncthreads();

    const int m     = lane & 15;
    const int koff  = (lane < 16) ? 0 : 2;
    const int rbase = tileBase + ((lane < 16) ? 0 : 8);
    const int ntiles = dhid >> 4;

    for (int nt = 0; nt < ntiles; ++nt) {
        const int col = nt * 16 + m;
        v8f acc = {};
#pragma unroll
        for (int kk = 0; kk < DINPAD; kk += 4) {
            const int k0 = kk + koff;                       // even
            const v2f a = *(const v2f*)&sh[wave][m * DINPAD + k0];
            const v2f b = *(const v2f*)&Wp[((size_t)(k0 >> 1) * dhid + col) * 2];
            acc = __builtin_amdgcn_wmma_f32_16x16x4_f32(
                false, a, false, b, (short)0, acc, false, false);
        }
        const float bv = bias[col];
#pragma unroll
        for (int i = 0; i < 8; ++i)
            Y[(size_t)(rbase + i) * dhid + col] = acc[i] + bv;
    }
}

// ---------------------------------------------------------------------------
// BatchNorm helpers: per-column sum / sumsq, finalize to scale/shift, apply.
// ---------------------------------------------------------------------------
__global__ void zero_kernel(float* __restrict__ p, int n)
{
    for (int t = threadIdx.x; t < n; t += blockDim.x) p[t] = 0.0f;
}

__global__ void col_stats(const float* __restrict__ Y, int R, int C,
                          float* __restrict__ sums)
{
    const int c  = threadIdx.x;                    // blockDim.x == C
    const int r0 = blockIdx.x * 256;
    int rend = r0 + 256; if (rend > R) rend = R;
    float s = 0.f, s2 = 0.f;
    for (int r = r0; r < rend; ++r) {
        float v = Y[(size_t)r * C + c];
        s += v; s2 += v * v;
    }
    atomicAdd(&sums[c], s);
    atomicAdd(&sums[C + c], s2);
}

__global__ void finalize_bn(const float* __restrict__ sums, float Rinv, int C,
                            const float* __restrict__ g, const float* __restrict__ be,
                            float* __restrict__ ss)
{
    const int c = threadIdx.x;
    if (c >= C) return;
    const float m = sums[c] * Rinv;
    const float v = sums[C + c] * Rinv - m * m;
    const float a = g[c] * rsqrtf(v + EPS_BN);
    ss[c]     = a;
    ss[C + c] = be[c] - m * a;
}

__global__ __launch_bounds__(256) void bn_relu(
    float* __restrict__ Y, const float* __restrict__ ss, int C, long total)
{
    long t = (long)blockIdx.x * 256 + threadIdx.x;
    const long stride = (long)gridDim.x * 256;
    for (; t < total; t += stride) {
        const int c = (int)(t % C);
        Y[t] = fmaxf(Y[t] * ss[c] + ss[C + c], 0.0f);
    }
}

// Max over K neighbors of relu(bn(y2)), written into FC at a column offset.
__global__ __launch_bounds__(256) void maxagg(
    const float* __restrict__ Y2, const float* __restrict__ ss, int C,
    float* __restrict__ FC, int fcstride, int colOff)
{
    const int t = blockIdx.x * 256 + threadIdx.x;  // n*C + c
    const int n = t / C;
    const int c = t - n * C;
    const float a  = ss[c];
    const float sh = ss[C + c];
    float m = 0.0f;                                // relu output is >= 0
    for (int k = 0; k < KNN; ++k) {
        float v = Y2[(size_t)(n * KNN + k) * C + c] * a + sh;
        v = fmaxf(v, 0.0f);
        m = fmaxf(m, v);
    }
    FC[(size_t)n * fcstride + colOff + c] = m;
}

// Final 128 -> 1 matvec.
__global__ __launch_bounds__(256) void logits_kernel(
    const float* __restrict__ Y, const float* __restrict__ W2,
    const float* __restrict__ b2, float* __restrict__ out)
{
    const int n = blockIdx.x * 256 + threadIdx.x;
    float s = 0.f;
#pragma unroll
    for (int c = 0; c < 128; ++c) s += Y[(size_t)n * 128 + c] * W2[c];
    out[n] = s + b2[0];
}

// ---------------------------------------------------------------------------
// Launch
// ---------------------------------------------------------------------------
extern "C" void kernel_launch(void* const* d_in, const int* in_sizes, int n_in,
                              void* d_out, int out_size, void* d_ws, size_t ws_size,
                              hipStream_t stream)
{
    // setup_inputs() leaf order: x, batch, then params in dict insertion order.
    const float* x    = (const float*)d_in[0];
    const float* sn_g = (const float*)d_in[2];
    const float* sn_b = (const float*)d_in[3];
    const float* tn_g = (const float*)d_in[4];
    const float* tn_b = (const float*)d_in[5];
    const float* sW1 = (const float*)d_in[6],  *sb1 = (const float*)d_in[7];
    const float* sg1 = (const float*)d_in[8],  *sbe1 = (const float*)d_in[9];
    const float* sW2 = (const float*)d_in[10], *sb2 = (const float*)d_in[11];
    const float* sg2 = (const float*)d_in[12], *sbe2 = (const float*)d_in[13];
    const float* pW1 = (const float*)d_in[14], *pb1 = (const float*)d_in[15];
    const float* pg1 = (const float*)d_in[16], *pbe1 = (const float*)d_in[17];
    const float* pW2 = (const float*)d_in[18], *pb2 = (const float*)d_in[19];
    const float* pg2 = (const float*)d_in[20], *pbe2 = (const float*)d_in[21];
    const float* fW1 = (const float*)d_in[22], *fb1 = (const float*)d_in[23];
    const float* fg1 = (const float*)d_in[24], *fbe1 = (const float*)d_in[25];
    const float* fW2 = (const float*)d_in[26], *fb2 = (const float*)d_in[27];
    const float* fg2 = (const float*)d_in[28], *fbe2 = (const float*)d_in[29];
    const float* cW1 = (const float*)d_in[30], *cb1 = (const float*)d_in[31];
    const float* cg  = (const float*)d_in[32], *cbe = (const float*)d_in[33];
    const float* cW2 = (const float*)d_in[34], *cb2 = (const float*)d_in[35];
    float* out = (float*)d_out;

    // Workspace layout (floats).
    float* ws = (float*)d_ws;
    const size_t R_E = (size_t)NNODES * KNN;       // 163840 edges
    float* XP   = ws;                               // 8192*4
    int*   IDX  = (int*)(ws + 32768);               // 8192*20 ints
    float* SUM  = ws + 32768 + 163840;              // 256
    float* SS   = SUM + 256;                        // 256
    float* INS  = SS + 256;                         // 8 (padded to 256)
    float* WP   = INS + 256;                        // 128*128 max padded weights
    float* FC   = WP + 16384;                       // 8192*128
    float* Ybuf = FC + (size_t)NNODES * 128;        // 163840*64 max
    float* Y2   = Ybuf + R_E * 64;                  // 163840*64 max
    (void)ws_size; (void)n_in; (void)in_sizes; (void)out_size;

    const int gemmBlocksE = (int)(R_E / 64);        // 4 tiles of 16 per block
    const int gemmBlocksN = NNODES / 64;

    // ---- input norms -> XP (xs = cols 0..2 of XP, xp = all 4 cols) ----
    input_stats<<<1, 256, 0, stream>>>(x, sn_g, sn_b, tn_g, tn_b, INS);
    apply_norm<<<(NNODES * 4) / 256, 256, 0, stream>>>(x, INS, XP);

    // ================= spatial stream (dim 3, din 6->8, hid 32) ==============
    knn_kernel<3><<<NNODES, 128, 0, stream>>>(XP, 4, IDX);
    pad_w<<<1, 256, 0, stream>>>(sW1, 6, 8, 32, WP);
    wmma_gemm<8><<<gemmBlocksE, 128, 0, stream>>>(XP, 4, 3, IDX, (const float*)nullptr,
                                                  6, WP, sb1, Ybuf, 32);
    zero_kernel<<<1, 256, 0, stream>>>(SUM, 64);
    col_stats<<<(int)(R_E / 256), 32, 0, stream>>>(Ybuf, (int)R_E, 32, SUM);
    finalize_bn<<<1, 32, 0, stream>>>(SUM, 1.0f / (float)R_E, 32, sg1, sbe1, SS);
    bn_relu<<<4096, 256, 0, stream>>>(Ybuf, SS, 32, (long)R_E * 32);
    pad_w<<<4, 256, 0, stream>>>(sW2, 32, 32, 32, WP);
    wmma_gemm<32><<<gemmBlocksE, 128, 0, stream>>>((const float*)nullptr, 0, 0,
                                                   (const int*)nullptr, Ybuf,
                                                   32, WP, sb2, Y2, 32);
    zero_kernel<<<1, 256, 0, stream>>>(SUM, 64);
    col_stats<<<(int)(R_E / 256), 32, 0, stream>>>(Y2, (int)R_E, 32, SUM);
    finalize_bn<<<1, 32, 0, stream>>>(SUM, 1.0f / (float)R_E, 32, sg2, sbe2, SS);
    maxagg<<<(NNODES * 32) / 256, 256, 0, stream>>>(Y2, SS, 32, FC, 128, 0);

    // ================= persistence stream (dim 4, din 8, hid 32) =============
    knn_kernel<4><<<NNODES, 128, 0, stream>>>(XP, 4, IDX);
    pad_w<<<1, 256, 0, stream>>>(pW1, 8, 8, 32, WP);
    wmma_gemm<8><<<gemmBlocksE, 128, 0, stream>>>(XP, 4, 4, IDX, (const float*)nullptr,
                                                  8, WP, pb1, Ybuf, 32);
    zero_kernel<<<1, 256, 0, stream>>>(SUM, 64);
    col_stats<<<(int)(R_E / 256), 32, 0, stream>>>(Ybuf, (int)R_E, 32, SUM);
    finalize_bn<<<1, 32, 0, stream>>>(SUM, 1.0f / (float)R_E, 32, pg1, pbe1, SS);
    bn_relu<<<4096, 256, 0, stream>>>(Ybuf, SS, 32, (long)R_E * 32);
    pad_w<<<4, 256, 0, stream>>>(pW2, 32, 32, 32, WP);
    wmma_gemm<32><<<gemmBlocksE, 128, 0, stream>>>((const float*)nullptr, 0, 0,
                                                   (const int*)nullptr, Ybuf,
                                                   32, WP, pb2, Y2, 32);
    zero_kernel<<<1, 256, 0, stream>>>(SUM, 64);
    col_stats<<<(int)(R_E / 256), 32, 0, stream>>>(Y2, (int)R_E, 32, SUM);
    finalize_bn<<<1, 32, 0, stream>>>(SUM, 1.0f / (float)R_E, 32, pg2, pbe2, SS);
    maxagg<<<(NNODES * 32) / 256, 256, 0, stream>>>(Y2, SS, 32, FC, 128, 32);

    // ================= fusion stream (dim 64 = FC cols 0..63) ================
    knn_kernel<64><<<NNODES, 128, 0, stream>>>(FC, 128, IDX);
    pad_w<<<32, 256, 0, stream>>>(fW1, 128, 128, 64, WP);
    wmma_gemm<128><<<gemmBlocksE, 128, 0, stream>>>(FC, 128, 64, IDX, (const float*)nullptr,
                                                    128, WP, fb1, Ybuf, 64);
    zero_kernel<<<1, 256, 0, stream>>>(SUM, 128);
    col_stats<<<(int)(R_E / 256), 64, 0, stream>>>(Ybuf, (int)R_E, 64, SUM);
    finalize_bn<<<1, 64, 0, stream>>>(SUM, 1.0f / (float)R_E, 64, fg1, fbe1, SS);
    bn_relu<<<4096, 256, 0, stream>>>(Ybuf, SS, 64, (long)R_E * 64);
    pad_w<<<16, 256, 0, stream>>>(fW2, 64, 64, 64, WP);
    wmma_gemm<64><<<gemmBlocksE, 128, 0, stream>>>((const float*)nullptr, 0, 0,
                                                   (const int*)nullptr, Ybuf,
                                                   64, WP, fb2, Y2, 64);
    zero_kernel<<<1, 256, 0, stream>>>(SUM, 128);
    col_stats<<<(int)(R_E / 256), 64, 0, stream>>>(Y2, (int)R_E, 64, SUM);
    finalize_bn<<<1, 64, 0, stream>>>(SUM, 1.0f / (float)R_E, 64, fg2, fbe2, SS);
    maxagg<<<(NNODES * 64) / 256, 256, 0, stream>>>(Y2, SS, 64, FC, 128, 64);

    // ================= classifier ============================================
    pad_w<<<64, 256, 0, stream>>>(cW1, 128, 128, 128, WP);
    wmma_gemm<128><<<gemmBlocksN, 128, 0, stream>>>((const float*)nullptr, 0, 0,
                                                    (const int*)nullptr, FC,
                                                    128, WP, cb1, Ybuf, 128);
    zero_kernel<<<1, 256, 0, stream>>>(SUM, 256);
    col_stats<<<NNODES / 256, 128, 0, stream>>>(Ybuf, NNODES, 128, SUM);
    finalize_bn<<<1, 128, 0, stream>>>(SUM, 1.0f / (float)NNODES, 128, cg, cbe, SS);
    bn_relu<<<4096, 256, 0, stream>>>(Ybuf, SS, 128, (long)NNODES * 128);
    logits_kernel<<<NNODES / 256, 256, 0, stream>>>(Ybuf, cW2, cb2, out);
}